// GAT_84817014161573
// MI455X (gfx1250) — compile-verified
//
#include <hip/hip_runtime.h>

// ---------------------------------------------------------------------------
// GAT (2-layer) for MI455X / gfx1250.
// N=50000 nodes, E=1.6M edges (+N self-loops), F_IN=256,
// layer1: 8 heads x 8 dims (concat 64), layer2: 1 head -> 16 classes.
// fp32 throughout (reference precision); GEMMs on the WMMA f32 path.
// ---------------------------------------------------------------------------

typedef __attribute__((ext_vector_type(2))) float v2f;
typedef __attribute__((ext_vector_type(8))) float v8f;

// ---- monotone float<->uint order encoding for atomicMax on floats ----------
__device__ __forceinline__ unsigned f2ord(float f) {
    unsigned u = __float_as_uint(f);
    return (u & 0x80000000u) ? ~u : (u | 0x80000000u);
}
__device__ __forceinline__ float ord2f(unsigned u) {
    return (u & 0x80000000u) ? __uint_as_float(u & 0x7fffffffu)
                             : __uint_as_float(~u);
}
__device__ __forceinline__ float lrelu(float x) { return x > 0.f ? x : 0.2f * x; }

// ---------------------------------------------------------------------------
// WMMA f32 GEMM: H[M x NOUT] = X[M x K] @ W[K x NOUT], row-major, fp32.
// Compile-time shape; one wave computes a 16 x (16*CTILES) strip so the A
// fragment is loaded once per K-step and fed to CTILES WMMAs.
// 8 waves per 256-thread block, one row tile each (wave-uniform guard keeps
// EXEC all-1s across every WMMA, per ISA 7.12 restrictions).
//
// V_WMMA_F32_16X16X4_F32 operand layout (wave32):
//   A (16x4):  lanes 0-15 = M rows; half-wave picks K{0,1} vs K{2,3};
//              per lane a contiguous float2 -> one global_load_b64.
//   B (4x16):  lane%16 = N col; VGPR0 = row K=2*half, VGPR1 = row K=2*half+1.
//   C/D:       VGPR r -> M = r + 8*half, N = lane%16.
// ---------------------------------------------------------------------------
template <int K, int NOUT, int CTILES>
__launch_bounds__(256)
__global__ void gat_gemm_wmma(const float* __restrict__ X,
                              const float* __restrict__ W,
                              float* __restrict__ H, int ntiles) {
    const int wave = threadIdx.x >> 5;
    const int tile = blockIdx.x * 8 + wave;      // row tile (16 rows)
    if (tile >= ntiles) return;                  // wave-uniform: EXEC stays ~0
    const int lane = threadIdx.x & 31;
    const int half = lane >> 4;                  // 0 or 1
    const int lid  = lane & 15;
    const int row0 = tile * 16;

    v8f acc[CTILES];
#pragma unroll
    for (int t = 0; t < CTILES; ++t)
        acc[t] = (v8f){0.f, 0.f, 0.f, 0.f, 0.f, 0.f, 0.f, 0.f};

    // per-lane A pointer: row (row0+lid), K offset 2*half (8B aligned)
    const float* __restrict__ xrow = X + (size_t)(row0 + lid) * K + 2 * half;

#pragma unroll 4
    for (int k0 = 0; k0 < K; k0 += 4) {
        const v2f a = *(const v2f*)(xrow + k0);  // A[K=2*half .. 2*half+1]
        const float* __restrict__ w0 = W + (size_t)(k0 + 2 * half) * NOUT + lid;
#pragma unroll
        for (int t = 0; t < CTILES; ++t) {
            v2f b;
            b.x = w0[t * 16];                    // B row K=2*half
            b.y = w0[t * 16 + NOUT];             // B row K=2*half+1
            acc[t] = __builtin_amdgcn_wmma_f32_16x16x4_f32(
                         /*neg_a=*/false, a, /*neg_b=*/false, b,
                         /*c_mod=*/(short)0, acc[t],
                         /*reuse_a=*/false, /*reuse_b=*/false);
        }
    }

#pragma unroll
    for (int t = 0; t < CTILES; ++t)
#pragma unroll
        for (int r = 0; r < 8; ++r)
            H[(size_t)(row0 + r + 8 * half) * NOUT + t * 16 + lid] = acc[t][r];
}

// ---------------------------------------------------------------------------
// Layer 1 node kernels (HEADS=8, HID=8, 64 channels)
// ---------------------------------------------------------------------------

// per (node, head): attention logits + self-loop logit seeds the segment max
__global__ void node_logits1(const float* __restrict__ h1,
                             const float* __restrict__ a_src,
                             const float* __restrict__ a_dst,
                             float* __restrict__ als, float* __restrict__ ald,
                             unsigned* __restrict__ menc, int N) {
    int t = blockIdx.x * blockDim.x + threadIdx.x;
    if (t >= N * 8) return;
    int n = t >> 3, h = t & 7;
    const float* hp = h1 + (size_t)n * 64 + h * 8;
    float s = 0.f, d = 0.f;
#pragma unroll
    for (int c = 0; c < 8; ++c) {
        float v = hp[c];
        s += v * a_src[h * 8 + c];
        d += v * a_dst[h * 8 + c];
    }
    als[t] = s;
    ald[t] = d;
    menc[t] = f2ord(lrelu(s + d));   // self-loop (src==dst==n)
}

// per edge: segment max of leaky-relu logits into dst (ordered-uint atomicMax)
__global__ void edge_max1(const int* __restrict__ src, const int* __restrict__ dst,
                          const float* __restrict__ als, const float* __restrict__ ald,
                          unsigned* __restrict__ menc, int E) {
    int e = blockIdx.x * blockDim.x + threadIdx.x;
    if (e >= E) return;
    int s = src[e], d = dst[e];
#pragma unroll
    for (int h = 0; h < 8; ++h) {
        float v = lrelu(als[s * 8 + h] + ald[d * 8 + h]);
        atomicMax(&menc[d * 8 + h], f2ord(v));
    }
}

// per (node, head): decode max, seed den & out with the self-loop contribution
__global__ void node_selfinit1(const float* __restrict__ h1,
                               const float* __restrict__ als,
                               const float* __restrict__ ald,
                               unsigned* __restrict__ menc,   // aliases mfl
                               float* __restrict__ mfl,
                               float* __restrict__ den, float* __restrict__ out1,
                               int N) {
    int t = blockIdx.x * blockDim.x + threadIdx.x;
    if (t >= N * 8) return;
    int n = t >> 3, h = t & 7;
    float m = ord2f(menc[t]);
    mfl[t] = m;                       // same slot, read-then-write: safe
    float ex = __expf(lrelu(als[t] + ald[t]) - m);
    den[t] = ex;
    const float* hp = h1 + (size_t)n * 64 + h * 8;
    float* op = out1 + (size_t)n * 64 + h * 8;
#pragma unroll
    for (int c = 0; c < 8; ++c) op[c] = hp[c] * ex;
}

// per (edge, head): unnormalized softmax-weighted scatter (den + features)
__global__ void edge_agg1(const int* __restrict__ src, const int* __restrict__ dst,
                          const float* __restrict__ als, const float* __restrict__ ald,
                          const float* __restrict__ mfl, const float* __restrict__ h1,
                          float* __restrict__ den, float* __restrict__ out1, int E) {
    int t = blockIdx.x * blockDim.x + threadIdx.x;
    if (t >= E * 8) return;
    int e = t >> 3, h = t & 7;
    int s = src[e], d = dst[e];
    float v = lrelu(als[s * 8 + h] + ald[d * 8 + h]);
    float ex = __expf(v - mfl[d * 8 + h]);
    atomicAdd(&den[d * 8 + h], ex);
    const float* hp = h1 + (size_t)s * 64 + h * 8;
    float* op = out1 + (size_t)d * 64 + h * 8;
#pragma unroll
    for (int c = 0; c < 8; ++c) atomicAdd(&op[c], hp[c] * ex);
}

// per (node, channel): normalize, +bias, ELU
__global__ void node_finish1(const float* __restrict__ out1,
                             const float* __restrict__ den,
                             const float* __restrict__ b1,
                             float* __restrict__ hmid, int N) {
    int t = blockIdx.x * blockDim.x + threadIdx.x;
    if (t >= N * 64) return;
    int n = t >> 6, c = t & 63;
    float v = out1[t] / den[n * 8 + (c >> 3)] + b1[c];
    hmid[t] = v > 0.f ? v : (__expf(v) - 1.f);   // ELU(alpha=1)
}

// ---------------------------------------------------------------------------
// Layer 2 kernels (1 head, 16 channels)
// ---------------------------------------------------------------------------
__global__ void node_logits2(const float* __restrict__ h2,
                             const float* __restrict__ a_src,
                             const float* __restrict__ a_dst,
                             float* __restrict__ als, float* __restrict__ ald,
                             unsigned* __restrict__ menc, int N) {
    int n = blockIdx.x * blockDim.x + threadIdx.x;
    if (n >= N) return;
    const float* hp = h2 + (size_t)n * 16;
    float s = 0.f, d = 0.f;
#pragma unroll
    for (int c = 0; c < 16; ++c) {
        float v = hp[c];
        s += v * a_src[c];
        d += v * a_dst[c];
    }
    als[n] = s;
    ald[n] = d;
    menc[n] = f2ord(lrelu(s + d));
}

__global__ void edge_max2(const int* __restrict__ src, const int* __restrict__ dst,
                          const float* __restrict__ als, const float* __restrict__ ald,
                          unsigned* __restrict__ menc, int E) {
    int e = blockIdx.x * blockDim.x + threadIdx.x;
    if (e >= E) return;
    float v = lrelu(als[src[e]] + ald[dst[e]]);
    atomicMax(&menc[dst[e]], f2ord(v));
}

__global__ void node_selfinit2(const float* __restrict__ h2,
                               const float* __restrict__ als,
                               const float* __restrict__ ald,
                               unsigned* __restrict__ menc, float* __restrict__ mfl,
                               float* __restrict__ den, float* __restrict__ out2,
                               int N) {
    int n = blockIdx.x * blockDim.x + threadIdx.x;
    if (n >= N) return;
    float m = ord2f(menc[n]);
    mfl[n] = m;
    float ex = __expf(lrelu(als[n] + ald[n]) - m);
    den[n] = ex;
    const float* hp = h2 + (size_t)n * 16;
    float* op = out2 + (size_t)n * 16;
#pragma unroll
    for (int c = 0; c < 16; ++c) op[c] = hp[c] * ex;
}

__global__ void edge_agg2(const int* __restrict__ src, const int* __restrict__ dst,
                          const float* __restrict__ als, const float* __restrict__ ald,
                          const float* __restrict__ mfl, const float* __restrict__ h2,
                          float* __restrict__ den, float* __restrict__ out2, int E) {
    int e = blockIdx.x * blockDim.x + threadIdx.x;
    if (e >= E) return;
    int s = src[e], d = dst[e];
    float v = lrelu(als[s] + ald[d]);
    float ex = __expf(v - mfl[d]);
    atomicAdd(&den[d], ex);
    const float* hp = h2 + (size_t)s * 16;
    float* op = out2 + (size_t)d * 16;
#pragma unroll
    for (int c = 0; c < 16; ++c) atomicAdd(&op[c], hp[c] * ex);
}

// per node: normalize, +bias, log_softmax over 16 classes
__global__ void node_logsoftmax(const float* __restrict__ out2,
                                const float* __restrict__ den,
                                const float* __restrict__ b2,
                                float* __restrict__ out, int N) {
    int n = blockIdx.x * blockDim.x + threadIdx.x;
    if (n >= N) return;
    float o[16];
    float inv = 1.f / den[n];
    float mx = -3.402823466e38f;
#pragma unroll
    for (int c = 0; c < 16; ++c) {
        o[c] = out2[(size_t)n * 16 + c] * inv + b2[c];
        mx = fmaxf(mx, o[c]);
    }
    float sum = 0.f;
#pragma unroll
    for (int c = 0; c < 16; ++c) sum += __expf(o[c] - mx);
    float lse = mx + __logf(sum);
#pragma unroll
    for (int c = 0; c < 16; ++c) out[(size_t)n * 16 + c] = o[c] - lse;
}

// ---------------------------------------------------------------------------
extern "C" void kernel_launch(void* const* d_in, const int* in_sizes, int n_in,
                              void* d_out, int out_size, void* d_ws, size_t ws_size,
                              hipStream_t stream) {
    const float* x      = (const float*)d_in[0];
    const int*   ei     = (const int*)  d_in[1];
    const float* W1     = (const float*)d_in[2];
    const float* a_src1 = (const float*)d_in[3];
    const float* a_dst1 = (const float*)d_in[4];
    const float* b1     = (const float*)d_in[5];
    const float* W2     = (const float*)d_in[6];
    const float* a_src2 = (const float*)d_in[7];
    const float* a_dst2 = (const float*)d_in[8];
    const float* b2     = (const float*)d_in[9];

    const int N = in_sizes[0] / 256;   // 50000 (multiple of 16 -> no GEMM guards)
    const int E = in_sizes[1] / 2;     // 1,600,000
    const int* src = ei;
    const int* dst = ei + E;

    // workspace carve-up (all fp32; ~52 MB total)
    float* ws   = (float*)d_ws;
    float* h1   = ws;  ws += (size_t)N * 64;
    float* als1 = ws;  ws += (size_t)N * 8;
    float* ald1 = ws;  ws += (size_t)N * 8;
    float* m1   = ws;  ws += (size_t)N * 8;   // uint-encoded during max pass
    float* den1 = ws;  ws += (size_t)N * 8;
    float* out1 = ws;  ws += (size_t)N * 64;
    float* hmid = ws;  ws += (size_t)N * 64;
    float* h2   = ws;  ws += (size_t)N * 16;
    float* als2 = ws;  ws += (size_t)N;
    float* ald2 = ws;  ws += (size_t)N;
    float* m2   = ws;  ws += (size_t)N;
    float* den2 = ws;  ws += (size_t)N;
    float* out2 = ws;  ws += (size_t)N * 16;

    const int TB = 256;                       // 8 wave32 waves per block
    const int ntiles = N / 16;                // 3125 row tiles
    const int gblk = (ntiles + 7) / 8;        // 8 row tiles per block

    // ---- layer 1 ----
    gat_gemm_wmma<256, 64, 4><<<gblk, TB, 0, stream>>>(x, W1, h1, ntiles);
    node_logits1<<<(N * 8 + TB - 1) / TB, TB, 0, stream>>>(
        h1, a_src1, a_dst1, als1, ald1, (unsigned*)m1, N);
    edge_max1<<<(E + TB - 1) / TB, TB, 0, stream>>>(src, dst, als1, ald1,
                                                    (unsigned*)m1, E);
    node_selfinit1<<<(N * 8 + TB - 1) / TB, TB, 0, stream>>>(
        h1, als1, ald1, (unsigned*)m1, m1, den1, out1, N);
    edge_agg1<<<((size_t)E * 8 + TB - 1) / TB, TB, 0, stream>>>(
        src, dst, als1, ald1, m1, h1, den1, out1, E);
    node_finish1<<<(N * 64 + TB - 1) / TB, TB, 0, stream>>>(out1, den1, b1, hmid, N);

    // ---- layer 2 ----
    gat_gemm_wmma<64, 16, 1><<<gblk, TB, 0, stream>>>(hmid, W2, h2, ntiles);
    node_logits2<<<(N + TB - 1) / TB, TB, 0, stream>>>(
        h2, a_src2, a_dst2, als2, ald2, (unsigned*)m2, N);
    edge_max2<<<(E + TB - 1) / TB, TB, 0, stream>>>(src, dst, als2, ald2,
                                                    (unsigned*)m2, E);
    node_selfinit2<<<(N + TB - 1) / TB, TB, 0, stream>>>(
        h2, als2, ald2, (unsigned*)m2, m2, den2, out2, N);
    edge_agg2<<<(E + TB - 1) / TB, TB, 0, stream>>>(
        src, dst, als2, ald2, m2, h2, den2, out2, E);
    node_logsoftmax<<<(N + TB - 1) / TB, TB, 0, stream>>>(
        out2, den2, b2, (float*)d_out, N);
}